// PhysChemLinearMix_49323404427796
// MI455X (gfx1250) — compile-verified
//
#include <hip/hip_runtime.h>
#include <float.h>
#include <math.h>

typedef __attribute__((ext_vector_type(2))) float v2f;
typedef __attribute__((ext_vector_type(4))) float v4f;
typedef __attribute__((ext_vector_type(8))) float v8f;

#define Bn 8192
#define Fn 32
#define Pn 512
#define Hn 256
#define XSTR 516   // 512 + 4 pad: 2064 B row stride (16B-aligned), 4-dword bank shift per row
#define HSTR 260

// ---- pre-kernel: Wt[p][h] = W1[h][p]  (512x256 floats, 512 KB, L2-resident) ----
__global__ void __launch_bounds__(256)
k_transpose_w1(const float* __restrict__ W1, float* __restrict__ Wt) {
    int idx = blockIdx.x * 256 + threadIdx.x;   // 0 .. P*H-1
    if (idx >= Pn * Hn) return;
    int p = idx >> 8;        // / Hn
    int h = idx & (Hn - 1);
    Wt[idx] = W1[h * Pn + p];                   // coalesced writes
}

// ---- fused kernel: one block = 16 samples ----
__global__ void __launch_bounds__(256)
k_fused(const float* __restrict__ phys, const float* __restrict__ ratios,
        const int* __restrict__ lengths, const float* __restrict__ gamma,
        const float* __restrict__ beta,  const float* __restrict__ Wt,
        const float* __restrict__ b1,    const float* __restrict__ W2,
        const float* __restrict__ b2,    float* __restrict__ out)
{
    __shared__ float xn[16 * XSTR];     // mix -> layernormed activations (A matrix)
    __shared__ float htile[16 * HSTR];  // relu(hidden) 16x256
    __shared__ float wmix[16 * Fn];     // normalized mix weights
    __shared__ float w2s[Hn];
    __shared__ float b1s[Hn];
    __shared__ float part[16 * 16];
    __shared__ int   lens[16];

    const int t    = threadIdx.x;
    const int lane = t & 31;
    const int wid  = t >> 5;
    const int b0   = blockIdx.x * 16;

    // ---- stage small params + per-sample normalized weights ----
    w2s[t] = W2[t];
    b1s[t] = b1[t];
    if (t < 16) {
        const int m = t;
        int L = lengths[b0 + m];
        if (L > Fn) L = Fn;
        if (L < 0)  L = 0;
        lens[m] = L;
        float s = 0.f;
        for (int f = 0; f < Fn; ++f) {
            float rv = (f < L) ? ratios[(size_t)(b0 + m) * Fn + f] : 0.f;
            wmix[m * Fn + f] = rv;
            s += rv;
        }
        const float inv = 1.f / (s + 1e-8f);
        for (int f = 0; f < Fn; ++f) wmix[m * Fn + f] *= inv;
    }
    __syncthreads();

    // ---- ragged weighted mix, b128 streaming ----
    // threads 0-127 handle samples 0..7, threads 128-255 handle samples 8..15;
    // each thread owns p-quad 4*(t&127) and accumulates a float4 per sample.
    {
        const int tq = t & 127;          // quad index -> p = 4*tq
        const int m0 = (t >> 7) * 8;     // sample base: 0 or 8
        for (int mm = 0; mm < 8; ++mm) {
            const int m = m0 + mm;
            const int L = lens[m];
            const v4f* __restrict__ base =
                (const v4f*)(phys + (size_t)(b0 + m) * Fn * Pn) + tq;
            v4f acc = {0.f, 0.f, 0.f, 0.f};
            for (int f = 0; f < L; ++f) {          // global_load_b128, 2KB/step per half-block
                const float wv = wmix[m * Fn + f];
                const v4f v = base[f * (Pn / 4)];
                acc.x += wv * v.x;
                acc.y += wv * v.y;
                acc.z += wv * v.z;
                acc.w += wv * v.w;
            }
            *(v4f*)(&xn[m * XSTR + 4 * tq]) = acc; // ds_store_b128 (16B aligned)
        }
    }
    __syncthreads();

    // ---- LayerNorm: wave wid handles samples 2*wid, 2*wid+1 ----
    for (int i = 0; i < 2; ++i) {
        const int m = 2 * wid + i;
        float s = 0.f, s2 = 0.f;
        for (int j = 0; j < 16; ++j) {
            const float v = xn[m * XSTR + lane + 32 * j];
            s += v; s2 += v * v;
        }
        for (int off = 16; off > 0; off >>= 1) {
            s  += __shfl_xor(s,  off, 32);
            s2 += __shfl_xor(s2, off, 32);
        }
        const float mu   = s  * (1.f / Pn);
        const float var  = s2 * (1.f / Pn) - mu * mu;
        const float rstd = rsqrtf(var + 1e-5f);
        for (int j = 0; j < 16; ++j) {
            const int p = lane + 32 * j;
            const float v = xn[m * XSTR + p];
            xn[m * XSTR + p] = (v - mu) * rstd * gamma[p] + beta[p];
        }
    }
    __syncthreads();

    // ---- GEMM 16x256 = xn(16x512) @ W1^T via V_WMMA_F32_16X16X4_F32 ----
    // A 16x4 layout: lanes 0-15 -> M=lane, K={k0,k0+1}; lanes 16-31 -> K={k0+2,k0+3}
    const int nlo  = lane & 15;
    const int koff = (lane >> 4) * 2;
    const int mbase = (lane >> 4) * 8;      // C/D: vgpr g holds M = g (+8 for hi lanes)
    for (int tt = 0; tt < 2; ++tt) {
        const int h0 = (wid + 8 * tt) * 16;
        v8f c = {0.f, 0.f, 0.f, 0.f, 0.f, 0.f, 0.f, 0.f};
        const float* __restrict__ wt = Wt + h0 + nlo;   // Wt[k][h], N-coalesced
        #pragma unroll 4
        for (int k0 = 0; k0 < Pn; k0 += 4) {
            v2f a, bf;
            a.x  = xn[nlo * XSTR + k0 + koff];
            a.y  = xn[nlo * XSTR + k0 + koff + 1];
            bf.x = wt[(size_t)(k0 + koff) * Hn];
            bf.y = wt[(size_t)(k0 + koff + 1) * Hn];
            c = __builtin_amdgcn_wmma_f32_16x16x4_f32(
                    /*neg_a=*/false, a, /*neg_b=*/false, bf,
                    /*c_mod=*/(short)0, c, /*reuse_a=*/false, /*reuse_b=*/false);
        }
        const float bb = b1s[h0 + nlo];
        #pragma unroll
        for (int g = 0; g < 8; ++g) {
            float v = c[g] + bb;
            v = v > 0.f ? v : 0.f;                       // ReLU
            htile[(mbase + g) * HSTR + h0 + nlo] = v;
        }
    }
    __syncthreads();

    // ---- final Linear(256 -> 1): y[m] = sum_h htile[m][h]*W2[h] + b2 ----
    {
        const int m = t >> 4, seg = t & 15;
        float s = 0.f;
        #pragma unroll
        for (int j = 0; j < 16; ++j) {
            const int h = seg * 16 + j;
            s += htile[m * HSTR + h] * w2s[h];
        }
        part[m * 16 + seg] = s;
    }
    __syncthreads();
    if (t < 16) {
        float y = b2[0];
        for (int seg = 0; seg < 16; ++seg) y += part[t * 16 + seg];
        if (isnan(y))      y = 0.f;                      // jnp.nan_to_num
        else if (isinf(y)) y = (y > 0.f) ? FLT_MAX : -FLT_MAX;
        out[b0 + t] = y;
    }
}

extern "C" void kernel_launch(void* const* d_in, const int* in_sizes, int n_in,
                              void* d_out, int out_size, void* d_ws, size_t ws_size,
                              hipStream_t stream) {
    (void)in_sizes; (void)n_in; (void)out_size; (void)ws_size;
    const float* phys    = (const float*)d_in[0];
    const float* ratios  = (const float*)d_in[1];
    const int*   lengths = (const int*)  d_in[2];
    const float* gamma   = (const float*)d_in[3];
    const float* beta    = (const float*)d_in[4];
    const float* W1      = (const float*)d_in[5];
    const float* b1      = (const float*)d_in[6];
    const float* W2      = (const float*)d_in[7];
    const float* b2      = (const float*)d_in[8];
    float*       out     = (float*)d_out;
    float*       Wt      = (float*)d_ws;          // 512*256 floats = 512 KB

    k_transpose_w1<<<(Pn * Hn + 255) / 256, 256, 0, stream>>>(W1, Wt);
    k_fused<<<Bn / 16, 256, 0, stream>>>(phys, ratios, lengths, gamma, beta,
                                         Wt, b1, W2, b2, out);
}